// GCMCLayer_58497454571672
// MI455X (gfx1250) — compile-verified
//
#include <hip/hip_runtime.h>
#include <hip/hip_bf16.h>
#include <math.h>

#define R  5
#define E  50000
#define NUu 20000
#define NIi 20000
#define NF 4
#define DN 16
#define DR 64
#define DO 64
#define TAU 0.5f

typedef __attribute__((ext_vector_type(2))) float v2f;
typedef __attribute__((ext_vector_type(8))) float v8f;
typedef __attribute__((address_space(3))) float lds_float;

// D = A(16x4) * B(4x16) + C, f32 WMMA
__device__ __forceinline__ v8f wmma4(v2f a, v2f b, v8f c) {
    return __builtin_amdgcn_wmma_f32_16x16x4_f32(false, a, false, b, (short)0, c, false, false);
}

__global__ void k_zero(float* __restrict__ p, int n) {
    int i = blockIdx.x * blockDim.x + threadIdx.x;
    if (i < n) p[i] = 0.0f;
}

// ---------------- per-edge blend + degree accumulation -------------------
__global__ void k_edge_prep(const float* __restrict__ user_h, const float* __restrict__ item_h,
                            const float* __restrict__ user_hsum, const float* __restrict__ item_hsum,
                            const float* __restrict__ review_feat, const float* __restrict__ prototypes,
                            const float* __restrict__ eta,
                            const int* __restrict__ edge_src, const int* __restrict__ edge_dst,
                            const int* __restrict__ kptr,
                            float* __restrict__ blended, float* __restrict__ norm_user,
                            float* __restrict__ norm_item)
{
    int ge = blockIdx.x * blockDim.x + threadIdx.x;
    if (ge >= R * E) return;
    int r = ge / E;
    int k = *kptr;
    int s = edge_src[ge], d = edge_dst[ge];

    const float* uh = user_h + ((size_t)k * R * NUu + (size_t)r * NUu + s) * DN;
    const float* ih = item_h + ((size_t)k * R * NIi + (size_t)r * NIi + d) * DN;
    float du = 0.f, di = 0.f, dot = 0.f;
    #pragma unroll
    for (int j = 0; j < DN; ++j) {
        float a = uh[j], b = ih[j];
        du += a * a; di += b * b; dot += a * b;
    }
    float nu_ = fmaxf(sqrtf(du), 1e-12f);
    float ni_ = fmaxf(sqrtf(di), 1e-12f);
    float sim_k = dot / (nu_ * ni_) / TAU;

    const float* ru = user_hsum + ((size_t)r * NUu + s) * NF * DN;
    const float* rc = item_hsum + ((size_t)r * NIi + d) * NF * DN;
    float sum_exp_sim = 0.f;
    #pragma unroll
    for (int f = 0; f < NF; ++f) {
        float acc = 0.f;
        #pragma unroll
        for (int j = 0; j < DN; ++j) acc += ru[f * DN + j] * rc[f * DN + j];
        sum_exp_sim += __expf(acc / TAU);
    }
    float exp_sim = __expf(sim_k) / sum_exp_sim;

    const float* rf = review_feat + (size_t)ge * NF * DR;
    float anchor_k = 0.f, sum_exp_anc = 0.f;
    for (int f = 0; f < NF; ++f) {
        float acc = 0.f;
        #pragma unroll 16
        for (int j = 0; j < DR; ++j) acc += rf[f * DR + j] * prototypes[f * DR + j];
        acc /= TAU;
        if (f == k) anchor_k = acc;
        sum_exp_anc += __expf(acc);
    }
    float exp_anchor = __expf(anchor_k) / sum_exp_anc;

    float g = 1.0f / (1.0f + __expf(-eta[ge]));
    float b = g * exp_anchor + (1.0f - g) * exp_sim;
    blended[ge] = b;
    atomicAdd(&norm_user[s], b);
    atomicAdd(&norm_item[d], b);
}

// ---------------- w = blended / sqrt(deg_u * deg_i), also int_dist out ---
__global__ void k_edge_w(const float* __restrict__ blended, const float* __restrict__ norm_user,
                         const float* __restrict__ norm_item,
                         const int* __restrict__ edge_src, const int* __restrict__ edge_dst,
                         float* __restrict__ w_ws, float* __restrict__ int_dist)
{
    int ge = blockIdx.x * blockDim.x + threadIdx.x;
    if (ge >= R * E) return;
    float nij = sqrtf(norm_user[edge_src[ge]] * norm_item[edge_dst[ge]]);
    float w = blended[ge] / nij;
    w_ws[ge] = w;
    int_dist[ge] = w;
}

// ---------------- node transform: hu = uh @ Wfwd^T, hi = ih @ Wrev^T -----
// One wave per 16-node tile; K=16 via 4 chained f32 WMMAs.
__global__ void k_node_transform(const float* __restrict__ user_h, const float* __restrict__ item_h,
                                 const float* __restrict__ node_w_fwd, const float* __restrict__ node_w_rev,
                                 const int* __restrict__ kptr,
                                 float* __restrict__ hu, float* __restrict__ hi)
{
    const int tiles_per_r = NUu / 16;                   // 1250
    int wave = (blockIdx.x * blockDim.x + threadIdx.x) >> 5;
    int lane = threadIdx.x & 31;
    if (wave >= R * tiles_per_r) return;
    int r  = wave / tiles_per_r;
    int n0 = (wave % tiles_per_r) * 16;
    int k  = *kptr;

    int mA = lane & 15;            // A row held by this lane
    int half = lane >> 4;
    int kh = half * 2;             // K sub-offset for this lane's A/B pair
    int n  = mA;                   // B/D column held by this lane

    const float* Au = user_h + ((size_t)k * R * NUu + (size_t)r * NUu + n0 + mA) * DN;
    const float* Ai = item_h + ((size_t)k * R * NIi + (size_t)r * NIi + n0 + mA) * DN;
    const float* Wf = node_w_fwd + (size_t)r * DN * DN; // (o,d) row-major; B[k][n]=Wf[n*DN+k]
    const float* Wr = node_w_rev + (size_t)r * DN * DN;

    v8f cf = {}; v8f cr = {};
    #pragma unroll
    for (int s = 0; s < 4; ++s) {
        int kk = 4 * s + kh;
        v2f au = { Au[kk], Au[kk + 1] };
        v2f ai = { Ai[kk], Ai[kk + 1] };
        v2f bf = { Wf[n * DN + kk], Wf[n * DN + kk + 1] };
        v2f br = { Wr[n * DN + kk], Wr[n * DN + kk + 1] };
        cf = wmma4(au, bf, cf);
        cr = wmma4(ai, br, cr);
    }
    #pragma unroll
    for (int v = 0; v < 8; ++v) {
        int m = v + 8 * half;
        hu[((size_t)r * NUu + n0 + m) * DN + n] = cf[v];
        hi[((size_t)r * NIi + n0 + m) * DN + n] = cr[v];
    }
}

// ------- fused review transform (WMMA, K=64) + weighted message scatter --
// One wave per 16-edge tile. rf_k tile staged memory->LDS with CDNA5 async
// copies (GLOBAL_LOAD_ASYNC_TO_LDS_B128, ASYNCcnt), per-wave wait, no barrier.
__global__ void __launch_bounds__(256)
k_messages(const float* __restrict__ review_feat,
           const float* __restrict__ review_w_fwd, const float* __restrict__ review_w_rev,
           const float* __restrict__ hu, const float* __restrict__ hi,
           const float* __restrict__ w_ws,
           const int* __restrict__ edge_src, const int* __restrict__ edge_dst,
           const int* __restrict__ kptr,
           float* __restrict__ user_msg, float* __restrict__ item_msg)
{
    __shared__ float lds[8 * 16 * DR];                  // 8 waves * 4KB = 32KB
    const int tiles_per_r = E / 16;                     // 3125
    int wave = (blockIdx.x * blockDim.x + threadIdx.x) >> 5;
    int wib  = threadIdx.x >> 5;
    int lane = threadIdx.x & 31;
    if (wave >= R * tiles_per_r) return;                // wave-uniform exit
    int k  = *kptr;
    float* tile = &lds[wib * 16 * DR];
    int r  = wave / tiles_per_r;
    int e0 = (wave % tiles_per_r) * 16;

    // raw LDS byte offset of this wave's tile (what async-to-LDS VDST expects)
    unsigned lds_base = (unsigned)(unsigned long long)(lds_float*)tile;

    // stage 16x64 f32 tile of rf_k: 256 float4s, 8 async b128 copies per lane
    #pragma unroll
    for (int it = 0; it < 8; ++it) {
        int i  = it * 32 + lane;                        // float4 index
        int m  = i >> 4;
        int c4 = (i & 15) * 4;
        unsigned long long gaddr = (unsigned long long)(review_feat +
            ((size_t)((size_t)r * E + e0 + m) * NF + k) * DR + c4);
        unsigned laddr = lds_base + (unsigned)((m * DR + c4) * 4);
        asm volatile("global_load_async_to_lds_b128 %0, %1, off"
                     :: "v"(laddr), "v"(gaddr) : "memory");
    }
    asm volatile("s_wait_asynccnt 0" ::: "memory");     // this wave's copies done

    int mA = lane & 15, half = lane >> 4, kh = half * 2, n = mA;
    const float* Wf = review_w_fwd + (size_t)r * DN * DR;  // (o=16, d=64); B[k][n]=Wf[n*DR+k]
    const float* Wr = review_w_rev + (size_t)r * DN * DR;

    v8f cf = {}; v8f cr = {};
    #pragma unroll
    for (int s = 0; s < 16; ++s) {
        int kk = 4 * s + kh;
        v2f a  = { tile[mA * DR + kk], tile[mA * DR + kk + 1] };
        v2f bf = { Wf[n * DR + kk], Wf[n * DR + kk + 1] };
        v2f br = { Wr[n * DR + kk], Wr[n * DR + kk + 1] };
        cf = wmma4(a, bf, cf);
        cr = wmma4(a, br, cr);
    }
    // scatter: lane owns column n of rows m = v + 8*half
    #pragma unroll
    for (int v = 0; v < 8; ++v) {
        int m  = v + 8 * half;
        int ge = r * E + e0 + m;
        float we = w_ws[ge];
        int s_ = edge_src[ge], d_ = edge_dst[ge];
        float fwd = (hu[((size_t)r * NUu + s_) * DN + n] + cf[v]) * we;
        float rev = (hi[((size_t)r * NIi + d_) * DN + n] + cr[v]) * we;
        atomicAdd(&item_msg[(size_t)d_ * DN + n], fwd);
        atomicAdd(&user_msg[(size_t)s_ * DN + n], rev);
    }
}

// ---------------- final FC: leaky_relu(msg) @ W^T + b (WMMA) -------------
__global__ void k_fc(const float* __restrict__ user_msg, const float* __restrict__ item_msg,
                     const float* __restrict__ ufc_w, const float* __restrict__ ufc_b,
                     const float* __restrict__ ifc_w, const float* __restrict__ ifc_b,
                     float* __restrict__ ufeat, float* __restrict__ ifeat)
{
    const int tiles = NUu / 16;                         // 1250 (NU == NI)
    int wave = (blockIdx.x * blockDim.x + threadIdx.x) >> 5;
    int lane = threadIdx.x & 31;
    if (wave >= 2 * tiles) return;
    bool is_item = wave >= tiles;
    int u0 = (is_item ? wave - tiles : wave) * 16;
    const float* msg = is_item ? item_msg : user_msg;
    const float* W   = is_item ? ifc_w : ufc_w;         // (DO=64, DN=16)
    const float* Bv  = is_item ? ifc_b : ufc_b;
    float* out       = is_item ? ifeat : ufeat;

    int mA = lane & 15, half = lane >> 4, kh = half * 2, n = mA;

    v2f afrag[4];
    #pragma unroll
    for (int s = 0; s < 4; ++s) {
        int kk = 4 * s + kh;
        float x0 = msg[(size_t)(u0 + mA) * DN + kk];
        float x1 = msg[(size_t)(u0 + mA) * DN + kk + 1];
        x0 = x0 > 0.f ? x0 : 0.1f * x0;                 // leaky_relu(0.1)
        x1 = x1 > 0.f ? x1 : 0.1f * x1;
        afrag[s] = (v2f){ x0, x1 };
    }
    #pragma unroll
    for (int nt = 0; nt < 4; ++nt) {                    // DO = 4 n-tiles of 16
        int o = nt * 16 + n;
        v8f c = {};
        #pragma unroll
        for (int s = 0; s < 4; ++s) {
            int kk = 4 * s + kh;
            v2f b = { W[o * DN + kk], W[o * DN + kk + 1] };
            c = wmma4(afrag[s], b, c);
        }
        float bias = Bv[o];
        #pragma unroll
        for (int v = 0; v < 8; ++v) {
            int m = v + 8 * half;
            out[(size_t)(u0 + m) * DO + o] = c[v] + bias;
        }
    }
}

extern "C" void kernel_launch(void* const* d_in, const int* in_sizes, int n_in,
                              void* d_out, int out_size, void* d_ws, size_t ws_size,
                              hipStream_t stream) {
    (void)in_sizes; (void)n_in; (void)out_size; (void)ws_size;
    const float* user_h       = (const float*)d_in[0];
    const float* item_h       = (const float*)d_in[1];
    const float* user_hsum    = (const float*)d_in[2];
    const float* item_hsum    = (const float*)d_in[3];
    const float* review_feat  = (const float*)d_in[4];
    const float* prototypes   = (const float*)d_in[5];
    const float* eta          = (const float*)d_in[6];
    const float* node_w_fwd   = (const float*)d_in[7];
    const float* review_w_fwd = (const float*)d_in[8];
    const float* node_w_rev   = (const float*)d_in[9];
    const float* review_w_rev = (const float*)d_in[10];
    const float* ufc_w        = (const float*)d_in[11];
    const float* ufc_b        = (const float*)d_in[12];
    const float* ifc_w        = (const float*)d_in[13];
    const float* ifc_b        = (const float*)d_in[14];
    const int*   edge_src     = (const int*)d_in[15];
    const int*   edge_dst     = (const int*)d_in[16];
    const int*   kptr         = (const int*)d_in[17];

    float* ws        = (float*)d_ws;
    float* blended   = ws;                 // R*E           = 250000
    float* w_ws      = ws +  250000;       // R*E           = 250000
    float* norm_user = ws +  500000;       // NU            =  20000
    float* norm_item = ws +  520000;       // NI            =  20000
    float* user_msg  = ws +  540000;       // NU*DN         = 320000
    float* item_msg  = ws +  860000;       // NI*DN         = 320000
    float* hu        = ws + 1180000;       // R*NU*DN       = 1600000
    float* hi        = ws + 2780000;       // R*NI*DN       = 1600000  (end 4380000 floats)

    float* ufeat    = (float*)d_out;           // NU*DO
    float* ifeat    = ufeat + (size_t)NUu * DO;
    float* int_dist = ifeat + (size_t)NIi * DO; // R*E

    // 1) zero accumulators (norm_user..item_msg are contiguous: 680000 floats)
    k_zero<<<(680000 + 255) / 256, 256, 0, stream>>>(ws + 500000, 680000);

    // 2) per-edge blend + degree sums
    k_edge_prep<<<(R * E + 255) / 256, 256, 0, stream>>>(
        user_h, item_h, user_hsum, item_hsum, review_feat, prototypes, eta,
        edge_src, edge_dst, kptr, blended, norm_user, norm_item);

    // 3) node transforms (WMMA): 5*1250 waves
    {
        int waves = R * (NUu / 16);
        int blocks = (waves * 32 + 255) / 256;
        k_node_transform<<<blocks, 256, 0, stream>>>(
            user_h, item_h, node_w_fwd, node_w_rev, kptr, hu, hi);
    }

    // 4) edge weights + int_dist output
    k_edge_w<<<(R * E + 255) / 256, 256, 0, stream>>>(
        blended, norm_user, norm_item, edge_src, edge_dst, w_ws, int_dist);

    // 5) fused review transform (WMMA + async-to-LDS) + message scatter
    {
        int waves = R * (E / 16);                       // 15625
        int blocks = (waves + 7) / 8;
        k_messages<<<blocks, 256, 0, stream>>>(
            review_feat, review_w_fwd, review_w_rev, hu, hi, w_ws,
            edge_src, edge_dst, kptr, user_msg, item_msg);
    }

    // 6) final FC (WMMA): 2*1250 waves
    {
        int waves = 2 * (NUu / 16);
        int blocks = (waves * 32 + 255) / 256;
        k_fc<<<blocks, 256, 0, stream>>>(
            user_msg, item_msg, ufc_w, ufc_b, ifc_w, ifc_b, ufeat, ifeat);
    }
}